// DynamicGraphConvLayer_86131274154648
// MI455X (gfx1250) — compile-verified
//
#include <hip/hip_runtime.h>

typedef __attribute__((ext_vector_type(2))) float v2f;
typedef __attribute__((ext_vector_type(8))) float v8f;

#define B_   4
#define C_   16
#define N_   1024
#define T_   24
#define K_   16
#define H_   32
#define OUT_ 64
#define CT_  (C_ * T_)   /* 384 */

// ---------------------------------------------------------------------------
// Pack Xf[b][n][c*T+t] = X[b][c][n][t]  (contiguous 384-dim rows for WMMA)
// ---------------------------------------------------------------------------
__global__ __launch_bounds__(256) void pack_kernel(const float* __restrict__ X,
                                                   float* __restrict__ Xfp) {
  const int i = blockIdx.x * blockDim.x + threadIdx.x;
  if (i >= B_ * N_ * CT_) return;
  const int k = i % CT_;
  const int n = (i / CT_) % N_;
  const int b = i / (CT_ * N_);
  const int cc = k / T_, tt = k % T_;
  Xfp[i] = X[((b * C_ + cc) * N_ + n) * T_ + tt];
}

// ---------------------------------------------------------------------------
// Squared norms: sq[b*N+n] = sum_k Xfp[b][n][k]^2   (one wave per row)
// ---------------------------------------------------------------------------
__global__ __launch_bounds__(256) void sq_kernel(const float* __restrict__ Xfp,
                                                 float* __restrict__ sq) {
  const int w = (blockIdx.x * blockDim.x + threadIdx.x) >> 5;
  const int lane = threadIdx.x & 31;
  if (w >= B_ * N_) return;
  const float* row = Xfp + (size_t)w * CT_;
  float s = 0.f;
  for (int k = lane; k < CT_; k += 32) { const float v = row[k]; s += v * v; }
  #pragma unroll
  for (int off = 16; off >= 1; off >>= 1) s += __shfl_xor(s, off, 32);
  if (lane == 0) sq[w] = s;
}

// ---------------------------------------------------------------------------
// Top-16 smallest per row of a (nrows x 1024) matrix, jax.lax.top_k tie order
// (ascending value, then ascending index). One wave per row.
// ---------------------------------------------------------------------------
__global__ __launch_bounds__(256) void topk_kernel(const float* __restrict__ mat,
                                                   int nrows,
                                                   int* __restrict__ outIdx) {
  const int w = (blockIdx.x * blockDim.x + threadIdx.x) >> 5;
  const int lane = threadIdx.x & 31;
  if (w >= nrows) return;
  const float* row = mat + (size_t)w * N_;
  float vals[32];
  #pragma unroll
  for (int jj = 0; jj < 32; ++jj) vals[jj] = row[jj * 32 + lane];
  float lastV = -3.4e38f; int lastJ = -1;
  for (int r = 0; r < K_; ++r) {
    float bv = 3.4e38f; int bj = 0x7fffffff;
    #pragma unroll
    for (int jj = 0; jj < 32; ++jj) {
      const float v = vals[jj];
      const int j = jj * 32 + lane;
      const bool elig   = (v > lastV) || (v == lastV && j > lastJ);
      const bool better = (v < bv)   || (v == bv   && j < bj);
      if (elig && better) { bv = v; bj = j; }
    }
    #pragma unroll
    for (int off = 16; off >= 1; off >>= 1) {
      const float ov = __shfl_xor(bv, off, 32);
      const int   oj = __shfl_xor(bj, off, 32);
      if (ov < bv || (ov == bv && oj < bj)) { bv = ov; bj = oj; }
    }
    if (lane == 0) outIdx[w * K_ + r] = bj;
    lastV = bv; lastJ = bj;
  }
}

// ---------------------------------------------------------------------------
// distF[b][i][j] = sq_i + sq_j - 2 * <Xf_i, Xf_j>  via f32 WMMA (16x16 tiles,
// K=384 in 96 steps of 4). One wave per tile; 8 waves / block.
// f32 A-layout: lanes 0-15 hold K=kk,kk+1 ; lanes 16-31 hold K=kk+2,kk+3.
// f32 B-layout mirrors it (B = Xf_jtile^T), so both fragments are float2 loads.
// ---------------------------------------------------------------------------
__global__ __launch_bounds__(256) void gram_kernel(const float* __restrict__ Xfp,
                                                   const float* __restrict__ sq,
                                                   float* __restrict__ distF) {
  const int wave = threadIdx.x >> 5;
  const int lane = threadIdx.x & 31;
  const int l16 = lane & 15, hi = lane >> 4;
  const int tile = blockIdx.x * 8 + wave;
  const int jt = tile & 63;
  const int it = (tile >> 6) & 63;
  const int b  = tile >> 12;

  const float* Ai = Xfp + (size_t)(b * N_ + it * 16) * CT_;
  const float* Bj = Xfp + (size_t)(b * N_ + jt * 16) * CT_;

  v8f c;
  #pragma unroll
  for (int v = 0; v < 8; ++v) c[v] = 0.f;

  for (int kk = 0; kk < CT_; kk += 4) {
    const v2f a = *(const v2f*)(Ai + (size_t)l16 * CT_ + kk + hi * 2);
    const v2f w = *(const v2f*)(Bj + (size_t)l16 * CT_ + kk + hi * 2);
    c = __builtin_amdgcn_wmma_f32_16x16x4_f32(false, a, false, w, (short)0, c,
                                              false, false);
  }

  const float sqj = sq[b * N_ + jt * 16 + l16];
  #pragma unroll
  for (int v = 0; v < 8; ++v) {
    const int i = it * 16 + hi * 8 + v;   // C layout: M = v (+8 for hi lanes)
    const float d = sq[b * N_ + i] + sqj - 2.f * c[v];
    distF[(size_t)(b * N_ + i) * N_ + jt * 16 + l16] = d;
  }
}

// ---------------------------------------------------------------------------
// Fused dual-branch edge-MLP + max-over-k. Rows ordered (t*16 + k) so a 16-row
// M-tile == one t with all 16 neighbors; max over k == max over M of one tile.
// One wave per t; block = 8 waves covers 8 t's; grid = B*N*3.
// ---------------------------------------------------------------------------
__global__ __launch_bounds__(256) void mlp_kernel(
    const float* __restrict__ X, const float* __restrict__ pos,
    const int* __restrict__ idxD, const int* __restrict__ idxF,
    const float* __restrict__ WD1, const float* __restrict__ bD1,
    const float* __restrict__ WD2, const float* __restrict__ bD2,
    const float* __restrict__ WF1, const float* __restrict__ bF1,
    const float* __restrict__ WF2, const float* __restrict__ bF2,
    float* __restrict__ out) {
  __shared__ float aT[8][16][37];   // per-wave A tile (<=36 feats), odd pad
  __shared__ float hT[8][16][33];   // per-wave hidden 16x32, odd pad

  const int wave = threadIdx.x >> 5;
  const int lane = threadIdx.x & 31;
  const int l16 = lane & 15, hi = lane >> 4;

  const int blk  = blockIdx.x;
  const int tblk = blk % 3;
  const int n    = (blk / 3) % N_;
  const int b    = blk / (3 * N_);
  const int t    = tblk * 8 + wave;

  float res[4] = {0.f, 0.f, 0.f, 0.f};

  for (int branch = 0; branch < 2; ++branch) {
    const int feat = (branch == 0) ? 32 : 36;
    const float* W1 = (branch == 0) ? WF1 : WD1;
    const float* B1 = (branch == 0) ? bF1 : bD1;
    const float* W2 = (branch == 0) ? WF2 : WD2;
    const float* B2 = (branch == 0) ? bF2 : bD2;

    // -- gather this wave's 16 x feat edge-feature tile into LDS --
    for (int e = lane; e < 16 * feat; e += 32) {
      const int k = e / feat;
      const int j = e % feat;
      float v;
      if (branch == 0) {  // F-branch: [center(16) | nbr-center(16)]
        const int idx = idxF[((b * N_) + n) * K_ + k];
        if (j < 16) v = X[((b * C_ + j) * N_ + n) * T_ + t];
        else        v = X[((b * C_ + (j - 16)) * N_ + idx) * T_ + t]
                      - X[((b * C_ + (j - 16)) * N_ + n  ) * T_ + t];
      } else {            // D-branch: [center(16)+pos(2) | diffs(16)+dpos(2)]
        const int idx = idxD[n * K_ + k];
        if (j < 18) {
          v = (j < 16) ? X[((b * C_ + j) * N_ + n) * T_ + t]
                       : pos[(j - 16) * N_ + n];
        } else {
          const int jj = j - 18;
          v = (jj < 16) ? (X[((b * C_ + jj) * N_ + idx) * T_ + t]
                         - X[((b * C_ + jj) * N_ + n  ) * T_ + t])
                        : (pos[(jj - 16) * N_ + idx] - pos[(jj - 16) * N_ + n]);
        }
      }
      aT[wave][k][j] = v;
    }
    __syncthreads();

    // -- layer 1: (16 x feat) @ (feat x 32) + bias, ReLU -> LDS hidden --
    for (int nt = 0; nt < 2; ++nt) {
      v8f c;
      const float bias = B1[nt * 16 + l16];
      #pragma unroll
      for (int v = 0; v < 8; ++v) c[v] = bias;
      for (int kk = 0; kk < feat; kk += 4) {
        v2f a, w;
        a.x = aT[wave][l16][kk + hi * 2];
        a.y = aT[wave][l16][kk + hi * 2 + 1];
        w.x = W1[(kk + hi * 2    ) * H_ + nt * 16 + l16];
        w.y = W1[(kk + hi * 2 + 1) * H_ + nt * 16 + l16];
        c = __builtin_amdgcn_wmma_f32_16x16x4_f32(false, a, false, w, (short)0,
                                                  c, false, false);
      }
      #pragma unroll
      for (int v = 0; v < 8; ++v)
        hT[wave][hi * 8 + v][nt * 16 + l16] = fmaxf(c[v], 0.f);
    }
    __syncthreads();

    // -- layer 2: (16 x 32) @ (32 x 64) + bias, ReLU, max over 16 k-rows --
    for (int nt = 0; nt < 4; ++nt) {
      v8f c;
      const float bias = B2[nt * 16 + l16];
      #pragma unroll
      for (int v = 0; v < 8; ++v) c[v] = bias;
      for (int kk = 0; kk < H_; kk += 4) {
        v2f a, w;
        a.x = hT[wave][l16][kk + hi * 2];
        a.y = hT[wave][l16][kk + hi * 2 + 1];
        w.x = W2[(kk + hi * 2    ) * OUT_ + nt * 16 + l16];
        w.y = W2[(kk + hi * 2 + 1) * OUT_ + nt * 16 + l16];
        c = __builtin_amdgcn_wmma_f32_16x16x4_f32(false, a, false, w, (short)0,
                                                  c, false, false);
      }
      float m = c[0];
      #pragma unroll
      for (int v = 1; v < 8; ++v) m = fmaxf(m, c[v]);   // max over M=v..v+8(hi)
      m = fmaxf(m, 0.f);                                // ReLU (max-monotone)
      m = fmaxf(m, __shfl_xor(m, 16, 32));              // combine lane halves
      res[nt] += m;
    }
    __syncthreads();
  }

  if (hi == 0) {
    #pragma unroll
    for (int nt = 0; nt < 4; ++nt) {
      const int o = nt * 16 + l16;
      out[((b * OUT_ + o) * N_ + n) * T_ + t] = res[nt];
    }
  }
}

// ---------------------------------------------------------------------------
extern "C" void kernel_launch(void* const* d_in, const int* in_sizes, int n_in,
                              void* d_out, int out_size, void* d_ws,
                              size_t ws_size, hipStream_t stream) {
  const float* X    = (const float*)d_in[0];
  const float* pos  = (const float*)d_in[1];
  const float* dist = (const float*)d_in[2];
  const float* WD1  = (const float*)d_in[3];
  const float* bD1  = (const float*)d_in[4];
  const float* WD2  = (const float*)d_in[5];
  const float* bD2  = (const float*)d_in[6];
  const float* WF1  = (const float*)d_in[7];
  const float* bF1  = (const float*)d_in[8];
  const float* WF2  = (const float*)d_in[9];
  const float* bF2  = (const float*)d_in[10];
  float* out = (float*)d_out;

  float* ws    = (float*)d_ws;
  float* Xfp   = ws;                                  // B*N*384 f32 (6 MB)
  float* distF = Xfp + (size_t)B_ * N_ * CT_;         // B*N*N  f32 (16 MB)
  float* sq    = distF + (size_t)B_ * N_ * N_;        // B*N    f32
  int*   idxD  = (int*)(sq + B_ * N_);                // N*K
  int*   idxF  = idxD + N_ * K_;                      // B*N*K

  pack_kernel<<<(B_ * N_ * CT_ + 255) / 256, 256, 0, stream>>>(X, Xfp);
  sq_kernel<<<(B_ * N_ * 32 + 255) / 256, 256, 0, stream>>>(Xfp, sq);
  topk_kernel<<<(N_ * 32 + 255) / 256, 256, 0, stream>>>(dist, N_, idxD);
  gram_kernel<<<(B_ * 64 * 64) / 8, 256, 0, stream>>>(Xfp, sq, distF);
  topk_kernel<<<(B_ * N_ * 32 + 255) / 256, 256, 0, stream>>>(distF, B_ * N_,
                                                              idxF);
  mlp_kernel<<<B_ * N_ * 3, 256, 0, stream>>>(X, pos, idxD, idxF, WD1, bD1,
                                              WD2, bD2, WF1, bF1, WF2, bF2,
                                              out);
}